// ODE_RNN_77051713290399
// MI455X (gfx1250) — compile-verified
//
#include <hip/hip_runtime.h>
#include <math.h>

// ---------------- problem constants (match reference) ----------------
#define HH   1024            // hidden
#define FF   2048
#define IIN  1024
#define TT   30
#define NSEG 8
#define SUB  4
#define BB   64              // batch == GRU sequence length
#define RR   (TT * NSEG)     // 240 independent GRU runs
#define G3H  (3 * HH)        // 3072

typedef __attribute__((ext_vector_type(2))) float v2f;
typedef __attribute__((ext_vector_type(4))) float v4f;
typedef __attribute__((ext_vector_type(8))) float v8f;

__device__ __forceinline__ float sigmoidf_(float x) {
    return 1.0f / (1.0f + expf(-x));
}

#define WMMA_F32(a, b, c) \
    __builtin_amdgcn_wmma_f32_16x16x4_f32(false, (a), false, (b), (short)0, (c), false, false)
#define LD2(p) (*(const v2f*)(p))

// ---------------------------------------------------------------------
// Generic fp32 WMMA GEMM (optionally batch-of-2 via blockIdx.z):
//   Cz[M][N] = Az[M][K] @ Wz[N][K]^T + biasz[N]
// Block = 128 threads (4 waves); each wave computes a 16(M) x 64(N) tile.
// grid = (N/256, M/16, 1 or 2). Requires N%256==0, M%16==0, K%16==0.
// Explicit depth-2 width-2 software pipeline: fragments for k-steps k and
// k+4 are held in named stage registers while k+8 / k+12 are prefetched,
// giving every gather load an 8-WMMA shadow and allowing partial
// s_wait_loadcnt waits instead of full barriers.
// ---------------------------------------------------------------------
__global__ void __launch_bounds__(128)
wmma_gemm_f32(const float* __restrict__ A0, const float* __restrict__ A1, int lda,
              const float* __restrict__ W0, const float* __restrict__ W1, int ldw,
              const float* __restrict__ bias0, const float* __restrict__ bias1,
              float* __restrict__ C0, float* __restrict__ C1, int ldc, int K)
{
    const float* A    = blockIdx.z ? A1    : A0;
    const float* W    = blockIdx.z ? W1    : W0;
    const float* bias = blockIdx.z ? bias1 : bias0;
    float*       C    = blockIdx.z ? C1    : C0;

    const int lane = threadIdx.x & 31;
    const int wave = threadIdx.x >> 5;
    const int half = lane >> 4;      // 0: lanes 0-15, 1: lanes 16-31
    const int lr   = lane & 15;
    const int row0 = blockIdx.y * 16;
    const int col0 = blockIdx.x * 256 + wave * 64;

    v8f acc0 = {}, acc1 = {}, acc2 = {}, acc3 = {};

    // A fragment (16x4, MxK): v0 = K(2*half), v1 = K(2*half+1), M = lr
    const float* ap  = A + (size_t)(row0 + lr) * lda + 2 * half;
    // B fragment (4x16, KxN): B[k][n] = W[n][k]; rows striped across lanes
    const float* wp0 = W + (size_t)(col0 +  0 + lr) * ldw + 2 * half;
    const float* wp1 = W + (size_t)(col0 + 16 + lr) * ldw + 2 * half;
    const float* wp2 = W + (size_t)(col0 + 32 + lr) * ldw + 2 * half;
    const float* wp3 = W + (size_t)(col0 + 48 + lr) * ldw + 2 * half;

    // pipeline prologue: stage X = fragments for k, stage Y = k+4
    v2f ax  = LD2(ap);     v2f ay  = LD2(ap  + 4);
    v2f bx0 = LD2(wp0);    v2f by0 = LD2(wp0 + 4);
    v2f bx1 = LD2(wp1);    v2f by1 = LD2(wp1 + 4);
    v2f bx2 = LD2(wp2);    v2f by2 = LD2(wp2 + 4);
    v2f bx3 = LD2(wp3);    v2f by3 = LD2(wp3 + 4);

    for (int k = 0; k <= K - 16; k += 8) {
        // prefetch k+8 while computing step k
        v2f ta  = LD2(ap  + k + 8);
        v2f t0  = LD2(wp0 + k + 8);
        v2f t1  = LD2(wp1 + k + 8);
        v2f t2  = LD2(wp2 + k + 8);
        v2f t3  = LD2(wp3 + k + 8);
        acc0 = WMMA_F32(ax, bx0, acc0);
        acc1 = WMMA_F32(ax, bx1, acc1);
        acc2 = WMMA_F32(ax, bx2, acc2);
        acc3 = WMMA_F32(ax, bx3, acc3);
        // prefetch k+12 while computing step k+4
        v2f ua  = LD2(ap  + k + 12);
        v2f u0  = LD2(wp0 + k + 12);
        v2f u1  = LD2(wp1 + k + 12);
        v2f u2  = LD2(wp2 + k + 12);
        v2f u3  = LD2(wp3 + k + 12);
        acc0 = WMMA_F32(ay, by0, acc0);
        acc1 = WMMA_F32(ay, by1, acc1);
        acc2 = WMMA_F32(ay, by2, acc2);
        acc3 = WMMA_F32(ay, by3, acc3);
        ax = ta; bx0 = t0; bx1 = t1; bx2 = t2; bx3 = t3;
        ay = ua; by0 = u0; by1 = u1; by2 = u2; by3 = u3;
    }
    // pipeline epilogue: final two k-steps (K-8, K-4)
    acc0 = WMMA_F32(ax, bx0, acc0);
    acc1 = WMMA_F32(ax, bx1, acc1);
    acc2 = WMMA_F32(ax, bx2, acc2);
    acc3 = WMMA_F32(ax, bx3, acc3);
    acc0 = WMMA_F32(ay, by0, acc0);
    acc1 = WMMA_F32(ay, by1, acc1);
    acc2 = WMMA_F32(ay, by2, acc2);
    acc3 = WMMA_F32(ay, by3, acc3);

    // C/D layout: VGPR v -> M = row0 + v + 8*half; N = col0 + sub*16 + lr
    const float bv0 = bias[col0 +  0 + lr];
    const float bv1 = bias[col0 + 16 + lr];
    const float bv2 = bias[col0 + 32 + lr];
    const float bv3 = bias[col0 + 48 + lr];
#pragma unroll
    for (int v = 0; v < 8; ++v) {
        const int m = row0 + v + 8 * half;
        float* crow = C + (size_t)m * ldc + col0;
        crow[ 0 + lr] = acc0[v] + bv0;
        crow[16 + lr] = acc1[v] + bv1;
        crow[32 + lr] = acc2[v] + bv2;
        crow[48 + lr] = acc3[v] + bv3;
    }
}

// ---------------------------------------------------------------------
// ODE MLP layer, wave-per-output dot product (M = 2 rows only).
// Out[row][n] = act( sum_k (In[row][k] + alpha*Kin[row][k]) * W[n][k] + bias[n] )
// 2*N waves; each lane reads float4 (wave reads 512B/iter, coalesced);
// shuffle reduction (wave32). Requires K % 128 == 0.
// ---------------------------------------------------------------------
__global__ void __launch_bounds__(256)
ode_layer(const float* __restrict__ In, const float* __restrict__ Kin, float alpha,
          const float* __restrict__ W, const float* __restrict__ bias,
          float* __restrict__ Out, int N, int K, int doTanh)
{
    const int wid  = blockIdx.x * (blockDim.x >> 5) + (threadIdx.x >> 5);
    const int lane = threadIdx.x & 31;
    const int n    = wid >> 1;
    const int row  = wid & 1;
    if (n >= N) return;

    const float* w  = W   + (size_t)n   * K;
    const float* h  = In  + (size_t)row * K;
    const float* kv = Kin + (size_t)row * K;

    float acc = 0.0f;
    for (int k = lane * 4; k < K; k += 128) {
        v4f hv = *(const v4f*)(h  + k);
        v4f kx = *(const v4f*)(kv + k);
        v4f wv = *(const v4f*)(w  + k);
        acc = fmaf(hv[0] + alpha * kx[0], wv[0], acc);
        acc = fmaf(hv[1] + alpha * kx[1], wv[1], acc);
        acc = fmaf(hv[2] + alpha * kx[2], wv[2], acc);
        acc = fmaf(hv[3] + alpha * kx[3], wv[3], acc);
    }
#pragma unroll
    for (int off = 16; off > 0; off >>= 1)
        acc += __shfl_xor(acc, off);
    if (lane == 0) {
        acc += bias[n];
        Out[(size_t)row * N + n] = doTanh ? tanhf(acc) : acc;
    }
}

// ---------------------------------------------------------------------
// Fused final layer of the k4 evaluation + RK4 state update:
//   k4 = u2 @ w3^T + b3   (per output element)
//   h  += dt/6 * (k1 + 2k2 + 2k3 + k4);  optional trajectory snapshot.
// ---------------------------------------------------------------------
__global__ void __launch_bounds__(256)
ode_layer_k4_combine(const float* __restrict__ In,
                     const float* __restrict__ W, const float* __restrict__ bias,
                     const float* __restrict__ k1, const float* __restrict__ k2,
                     const float* __restrict__ k3,
                     float* __restrict__ h, float* __restrict__ traj_slot,
                     float dt, int N, int K)
{
    const int wid  = blockIdx.x * (blockDim.x >> 5) + (threadIdx.x >> 5);
    const int lane = threadIdx.x & 31;
    const int n    = wid >> 1;
    const int row  = wid & 1;
    if (n >= N) return;

    const float* w = W  + (size_t)n   * K;
    const float* u = In + (size_t)row * K;

    float acc = 0.0f;
    for (int k = lane * 4; k < K; k += 128) {
        v4f uv = *(const v4f*)(u + k);
        v4f wv = *(const v4f*)(w + k);
        acc = fmaf(uv[0], wv[0], acc);
        acc = fmaf(uv[1], wv[1], acc);
        acc = fmaf(uv[2], wv[2], acc);
        acc = fmaf(uv[3], wv[3], acc);
    }
#pragma unroll
    for (int off = 16; off > 0; off >>= 1)
        acc += __shfl_xor(acc, off);
    if (lane == 0) {
        const size_t e = (size_t)row * N + n;
        const float k4v = acc + bias[n];
        const float v = h[e] + (dt / 6.0f) * (k1[e] + 2.0f * k2[e] + 2.0f * k3[e] + k4v);
        h[e] = v;
        if (traj_slot) traj_slot[e] = v;
    }
}

__global__ void zero_buf(float* __restrict__ p, int n) {
    const int i = blockIdx.x * blockDim.x + threadIdx.x;
    if (i < n) p[i] = 0.0f;
}

// X2[(t*BB + b)][k] = x[b][t][k]   (rows ordered so GEMM row = t*64+b)
__global__ void repack_x(const float* __restrict__ x, float* __restrict__ X2) {
    const int idx = blockIdx.x * blockDim.x + threadIdx.x;
    if (idx >= TT * BB * IIN) return;
    const int k  = idx & (IIN - 1);
    const int tb = idx >> 10;        // IIN == 1024
    const int t  = tb / BB;
    const int b  = tb % BB;
    X2[idx] = x[((size_t)b * TT + t) * IIN + k];
}

// H1[r][:] = traj[i][0][:], H2[r][:] = traj[i][1][:]  with r = t*8+i
__global__ void init_gru_states(const float* __restrict__ traj,
                                float* __restrict__ H1, float* __restrict__ H2)
{
    const int idx = blockIdx.x * blockDim.x + threadIdx.x;
    if (idx >= RR * HH) return;
    const int j = idx & (HH - 1);
    const int r = idx >> 10;
    const int i = r & 7;
    H1[idx] = traj[((size_t)i * 2 + 0) * HH + j];
    H2[idx] = traj[((size_t)i * 2 + 1) * HH + j];
}

// layer-0 GRU elementwise: H1 <- cell(gi0_row, gh0_row, H1)
__global__ void __launch_bounds__(256)
gru_ew1(const float* __restrict__ Gi0, const float* __restrict__ GH0,
        float* __restrict__ H1, int b)
{
    const int idx = blockIdx.x * blockDim.x + threadIdx.x;
    if (idx >= RR * HH) return;
    const int j = idx & (HH - 1);
    const int r = idx >> 10;
    const int t = r >> 3;
    const float* gi = Gi0 + ((size_t)t * BB + b) * G3H;   // shared over the 8 i's
    const float* gh = GH0 + (size_t)r * G3H;
    const float ir = gi[j], iz = gi[HH + j], in = gi[2 * HH + j];
    const float hr = gh[j], hz = gh[HH + j], hn = gh[2 * HH + j];
    const float rg = sigmoidf_(ir + hr);
    const float z  = sigmoidf_(iz + hz);
    const float n  = tanhf(in + rg * hn);
    H1[idx] = (1.0f - z) * n + z * H1[idx];
}

// layer-1 GRU elementwise: H2 <- cell(gi1_row, gh1_row, H2); write output
__global__ void __launch_bounds__(256)
gru_ew2(const float* __restrict__ GI1, const float* __restrict__ GH1,
        float* __restrict__ H2, float* __restrict__ out, int b)
{
    const int idx = blockIdx.x * blockDim.x + threadIdx.x;
    if (idx >= RR * HH) return;
    const int j = idx & (HH - 1);
    const int r = idx >> 10;
    const float* gi = GI1 + (size_t)r * G3H;
    const float* gh = GH1 + (size_t)r * G3H;
    const float ir = gi[j], iz = gi[HH + j], in = gi[2 * HH + j];
    const float hr = gh[j], hz = gh[HH + j], hn = gh[2 * HH + j];
    const float rg = sigmoidf_(ir + hr);
    const float z  = sigmoidf_(iz + hz);
    const float n  = tanhf(in + rg * hn);
    const float v  = (1.0f - z) * n + z * H2[idx];
    H2[idx] = v;
    out[((size_t)b * RR + r) * HH + j] = v;   // (B, 240, H), dim1 index = t*8+i
}

// ---------------------------------------------------------------------
extern "C" void kernel_launch(void* const* d_in, const int* in_sizes, int n_in,
                              void* d_out, int out_size, void* d_ws, size_t ws_size,
                              hipStream_t stream)
{
    const float* x   = (const float*)d_in[0];
    const float* w1  = (const float*)d_in[1];
    const float* b1  = (const float*)d_in[2];
    const float* w2  = (const float*)d_in[3];
    const float* b2  = (const float*)d_in[4];
    const float* w3  = (const float*)d_in[5];
    const float* b3  = (const float*)d_in[6];
    const float* wi0 = (const float*)d_in[7];
    const float* wh0 = (const float*)d_in[8];
    const float* bi0 = (const float*)d_in[9];
    const float* bh0 = (const float*)d_in[10];
    const float* wi1 = (const float*)d_in[11];
    const float* wh1 = (const float*)d_in[12];
    const float* bi1 = (const float*)d_in[13];
    const float* bh1 = (const float*)d_in[14];
    float* out = (float*)d_out;

    // ---- workspace carve-up (floats); total ~42.5 MB ----
    float* ws   = (float*)d_ws;
    float* X2   = ws;               ws += (size_t)TT * BB * IIN;   // 1.97M f
    float* Gi0  = ws;               ws += (size_t)TT * BB * G3H;   // 5.90M f
    float* H1   = ws;               ws += (size_t)RR * HH;
    float* H2   = ws;               ws += (size_t)RR * HH;
    float* GH0  = ws;               ws += (size_t)RR * G3H;
    float* GH1  = ws;               ws += (size_t)RR * G3H;
    float* GI1  = ws;               ws += (size_t)RR * G3H;
    float* traj = ws;               ws += (size_t)NSEG * 2 * HH;
    float* hode = ws;               ws += 2 * HH;
    float* u1   = ws;               ws += 2 * FF;
    float* u2   = ws;               ws += 2 * FF;
    float* k1   = ws;               ws += 2 * HH;
    float* k2   = ws;               ws += 2 * HH;
    float* k3   = ws;               ws += 2 * HH;
    (void)ws_size; (void)in_sizes; (void)n_in; (void)out_size;

    const float dt = (1.0f / NSEG) / SUB;   // 1/32

    // ---------------- phase 1: parallel input-gate GEMM (WMMA) ----------------
    repack_x<<<(TT * BB * IIN + 255) / 256, 256, 0, stream>>>(x, X2);
    // Gi0[t*64+b][n] = x[b][t] @ wi0^T + bi0   (1920 x 3072, K=1024)
    wmma_gemm_f32<<<dim3(G3H / 256, (TT * BB) / 16, 1), 128, 0, stream>>>(
        X2, X2, IIN, wi0, wi0, IIN, bi0, bi0, Gi0, Gi0, G3H, IIN);

    // ---------------- phase 2: ODE trajectory (computed once; t-invariant) ----
    zero_buf<<<(2 * HH + 255) / 256, 256, 0, stream>>>(hode, 2 * HH);
    for (int seg = 0; seg < NSEG; ++seg) {
        for (int sub = 0; sub < SUB; ++sub) {
            const float  alphas[3] = {0.0f, 0.5f * dt, 0.5f * dt};
            const float* kins[3]   = {hode, k1, k2};
            float*       kouts[3]  = {k1, k2, k3};
            // evals k1..k3 (three layers each); ode_f(h + alpha*k)
            for (int e = 0; e < 3; ++e) {
                ode_layer<<<FF / 4, 256, 0, stream>>>(hode, kins[e], alphas[e],
                                                      w1, b1, u1, FF, HH, 1);
                ode_layer<<<FF / 4, 256, 0, stream>>>(u1, u1, 0.0f,
                                                      w2, b2, u2, FF, FF, 1);
                ode_layer<<<HH / 4, 256, 0, stream>>>(u2, u2, 0.0f,
                                                      w3, b3, kouts[e], HH, FF, 0);
            }
            // eval k4 (input h + dt*k3), last layer fused with the RK4 update
            ode_layer<<<FF / 4, 256, 0, stream>>>(hode, k3, dt,
                                                  w1, b1, u1, FF, HH, 1);
            ode_layer<<<FF / 4, 256, 0, stream>>>(u1, u1, 0.0f,
                                                  w2, b2, u2, FF, FF, 1);
            float* slot = (sub == SUB - 1) ? (traj + (size_t)seg * 2 * HH) : nullptr;
            ode_layer_k4_combine<<<HH / 4, 256, 0, stream>>>(
                u2, w3, b3, k1, k2, k3, hode, slot, dt, HH, FF);
        }
    }

    // ---------------- phase 3: 240-way batched recurrent GRU ----------------
    init_gru_states<<<(RR * HH + 255) / 256, 256, 0, stream>>>(traj, H1, H2);

    const dim3 ggrid1(G3H / 256, RR / 16, 1);   // (12, 15, 1)
    const dim3 ggrid2(G3H / 256, RR / 16, 2);   // both recurrent-gate GEMMs at once
    const int  ew_blocks = (RR * HH + 255) / 256;
    for (int b = 0; b < BB; ++b) {
        // recurrent gates for both layers off the previous hidden states (z = 0/1)
        wmma_gemm_f32<<<ggrid2, 128, 0, stream>>>(H1, H2, HH, wh0, wh1, HH,
                                                  bh0, bh1, GH0, GH1, G3H, HH);
        // layer-0 cell update (uses hoisted Gi0 row for this (t, step))
        gru_ew1<<<ew_blocks, 256, 0, stream>>>(Gi0, GH0, H1, b);
        // layer-1 input gates from the *updated* H1
        wmma_gemm_f32<<<ggrid1, 128, 0, stream>>>(H1, H1, HH, wi1, wi1, HH,
                                                  bi1, bi1, GI1, GI1, G3H, HH);
        // layer-1 cell update + emit output
        gru_ew2<<<ew_blocks, 256, 0, stream>>>(GI1, GH1, H2, out, b);
    }
}